// NonMaxSuppression_32976758898717
// MI455X (gfx1250) — compile-verified
//
#include <hip/hip_runtime.h>
#include <stdint.h>

// Problem geometry (fixed by the reference harness).
#define NMS_N    2048          // boxes per batch (power of two -> bitonic sort)
#define NMS_NT   256           // threads per workgroup (8 wave32)
#define NMS_NW   (NMS_N / 32)  // suppression bitmask words
#define NMS_KMAX 256           // max K we stage in LDS (K=100 here)

typedef unsigned int v4u __attribute__((ext_vector_type(4)));
typedef int          v8i __attribute__((ext_vector_type(8)));
typedef int          v4i __attribute__((ext_vector_type(4)));

// float -> u32 key such that ascending u32 order == DESCENDING float order
__device__ __forceinline__ unsigned enc_desc(float s) {
    unsigned u   = __float_as_uint(s);
    unsigned asc = (u & 0x80000000u) ? ~u : (u ^ 0x80000000u);
    return ~asc;
}
__device__ __forceinline__ float dec_desc(unsigned key) {
    unsigned asc = ~key;
    unsigned u   = (asc & 0x80000000u) ? (asc ^ 0x80000000u) : ~asc;
    return __uint_as_float(u);
}

struct SMemT {
    unsigned long long keys[NMS_N];  // (desc-key << 32) | original index
    float bx1[NMS_N];
    float by1[NMS_N];
    float bx2[NMS_N];
    float by2[NMS_N];
    float scr[NMS_N];                // scores staging (TDM target), reused as area[]
    unsigned supp[NMS_NW];           // suppression bitmask, bit j
    int  sel[NMS_KMAX];              // kept original indices, rank order
    int  scnt;                       // #outputs passing score threshold (prefix)
};

__global__ __launch_bounds__(NMS_NT)
void nms_kernel(const float* __restrict__ boxes,
                const float* __restrict__ scores,
                const float* __restrict__ dIou,
                const float* __restrict__ dSth,
                int* __restrict__ out,
                int C, int K)
{
    __shared__ SMemT sm;
    const int tid = (int)threadIdx.x;
    const int p   = (int)blockIdx.x;        // p = b*C + c
    const int b   = p / C;
    const int c   = p % C;
    const int Kc  = (K < NMS_KMAX) ? K : NMS_KMAX;

    const float iou_thr = dIou[0];
    const float sthr    = dSth[0];
    const float* scores_p = scores + (size_t)p * NMS_N;
    const float4* boxes_p = (const float4*)(boxes + (size_t)b * NMS_N * 4);

    // ---- Stage scores into LDS via the Tensor Data Mover (async DMA) ----
#if __has_builtin(__builtin_amdgcn_tensor_load_to_lds) && __has_builtin(__builtin_amdgcn_s_wait_tensorcnt)
    if (tid < 32) {  // TDM is a per-wave op (EXEC ignored): issue from wave 0 only
        unsigned lds_scr = (unsigned)(uintptr_t)(&sm.scr[0]);            // generic LDS ptr low 32b = LDS offset
        unsigned long long ga = (unsigned long long)(uintptr_t)scores_p; // 57-bit global address
        v4u g0;
        g0.x = 1u;                                           // count=1, user descriptor
        g0.y = lds_scr;                                      // lds_addr
        g0.z = (unsigned)(ga & 0xFFFFFFFFu);                 // global_addr[31:0]
        g0.w = (unsigned)((ga >> 32) & 0x01FFFFFFu) | (2u << 30); // global_addr[56:32] | type=2
        v8i g1;
        g1[0] = (int)(2u << 16);                    // workgroup_mask=0, data_size=2 (4 bytes)
        g1[1] = (int)((unsigned)NMS_N << 16);       // tensor_dim0 = 2048 (low 16 in bits[31:16])
        g1[2] = (int)(1u << 16);                    // tensor_dim0 hi=0 | tensor_dim1 = 1
        g1[3] = (int)((unsigned)NMS_N << 16);       // tensor_dim1 hi=0 | tile_dim0 = 2048
        g1[4] = 1;                                  // tile_dim1 = 1, tile_dim2 = 0 (unused)
        g1[5] = NMS_N;                              // tensor_dim0_stride = 2048
        g1[6] = 0;
        g1[7] = 0;
        v4i gz4 = {0, 0, 0, 0};
        v8i gz8 = {0, 0, 0, 0, 0, 0, 0, 0};
        __builtin_amdgcn_tensor_load_to_lds(g0, g1, gz4, gz4, gz8, 0);
        __builtin_amdgcn_s_wait_tensorcnt(0);
    }
#else
    for (int i = tid; i < NMS_N; i += NMS_NT) sm.scr[i] = scores_p[i];
#endif
    if (tid == 0) sm.scnt = 0;
    for (int w = tid; w < NMS_NW; w += NMS_NT) sm.supp[w] = 0u;
    __syncthreads();

    // ---- Build sortable keys: descending score, stable by index ----
    for (int i = tid; i < NMS_N; i += NMS_NT)
        sm.keys[i] = ((unsigned long long)enc_desc(sm.scr[i]) << 32) | (unsigned)i;

    // ---- Bitonic sort (ascending u64 == descending score) ----
    for (unsigned k = 2; k <= NMS_N; k <<= 1) {
        for (unsigned j = k >> 1; j > 0; j >>= 1) {
            __syncthreads();
            for (unsigned i = (unsigned)tid; i < NMS_N; i += NMS_NT) {
                unsigned ixj = i ^ j;
                if (ixj > i) {
                    unsigned long long a  = sm.keys[i];
                    unsigned long long b2 = sm.keys[ixj];
                    bool up = ((i & k) == 0);
                    if ((a > b2) == up) { sm.keys[i] = b2; sm.keys[ixj] = a; }
                }
            }
        }
    }
    __syncthreads();

    // ---- Gather boxes in rank order, fix corners, compute areas ----
    float* area = sm.scr;  // scores staging is dead; reuse as area[]
    for (int r = tid; r < NMS_N; r += NMS_NT) {
        unsigned idx = (unsigned)(sm.keys[r] & 0xFFFFFFFFu);
        float4 bb = boxes_p[idx];
        float x1 = fminf(bb.x, bb.z), y1 = fminf(bb.y, bb.w);
        float x2 = fmaxf(bb.x, bb.z), y2 = fmaxf(bb.y, bb.w);
        sm.bx1[r] = x1; sm.by1[r] = y1; sm.bx2[r] = x2; sm.by2[r] = y2;
        area[r] = (x2 - x1) * (y2 - y1);
    }
    __syncthreads();

    // ---- Sequential suppress-scan (early exit at K kept) ----
    int nkept = 0;
    for (int k = 0; k < NMS_N; ++k) {
        bool kept = ((sm.supp[k >> 5] >> (k & 31)) & 1u) == 0u;  // uniform across block
        if (!kept) continue;
        if (tid == 0) {
            sm.sel[nkept] = (int)(sm.keys[k] & 0xFFFFFFFFu);
            float s = dec_desc((unsigned)(sm.keys[k] >> 32));
            if (s >= sthr) sm.scnt = nkept + 1;   // scores monotone desc -> prefix count
        }
        ++nkept;
        if (nkept >= Kc) break;                   // rank >= K can never be emitted

        float kx1 = sm.bx1[k], ky1 = sm.by1[k];
        float kx2 = sm.bx2[k], ky2 = sm.by2[k];
        float ka  = area[k];
        for (int base = 0; base < NMS_N; base += NMS_NT) {
            int j = base + tid;                   // lane l of a wave covers bit l of word j>>5
            bool sup = false;
            if (j > k) {
                float ix1 = fmaxf(kx1, sm.bx1[j]);
                float iy1 = fmaxf(ky1, sm.by1[j]);
                float ix2 = fminf(kx2, sm.bx2[j]);
                float iy2 = fminf(ky2, sm.by2[j]);
                float inter = fmaxf(ix2 - ix1, 0.f) * fmaxf(iy2 - iy1, 0.f);
                float iou   = inter / (ka + area[j] - inter);
                sup = iou > iou_thr;
            }
#if __has_builtin(__builtin_amdgcn_ballot_w32)
            unsigned m = (unsigned)__builtin_amdgcn_ballot_w32(sup);  // wave32: 1 ballot == 1 mask word
#else
            unsigned m = (unsigned)__ballot(sup);
#endif
            if ((tid & 31) == 0 && m) atomicOr(&sm.supp[(unsigned)j >> 5], m);
        }
        __syncthreads();
    }
    __syncthreads();

    // ---- Emit (b, c, idx) triples; -1 padding past scnt ----
    const int cnt = sm.scnt;
    for (int r = tid; r < K; r += NMS_NT) {
        long long o = ((long long)p * K + r) * 3;
        int v0 = -1, v1 = -1, v2 = -1;
        if (r < cnt) { v0 = b; v1 = c; v2 = sm.sel[r]; }
        out[o + 0] = v0;
        out[o + 1] = v1;
        out[o + 2] = v2;
    }
}

extern "C" void kernel_launch(void* const* d_in, const int* in_sizes, int n_in,
                              void* d_out, int out_size, void* d_ws, size_t ws_size,
                              hipStream_t stream) {
    (void)d_ws; (void)ws_size; (void)n_in;
    const float* boxes  = (const float*)d_in[0];
    const float* scores = (const float*)d_in[1];
    // d_in[2] = max_output_boxes_per_class (device int, = K)
    const float* dIou = (const float*)d_in[3];
    const float* dSth = (const float*)d_in[4];
    int* out = (int*)d_out;

    const int B = 4;                       // reference harness geometry
    const int N = in_sizes[0] / (B * 4);   // 2048
    const int C = in_sizes[1] / (B * N);   // 16
    const int K = out_size / (B * C * 3);  // 100
    (void)N;

    nms_kernel<<<dim3(B * C), dim3(NMS_NT), 0, stream>>>(boxes, scores, dIou, dSth, out, C, K);
}